// SKELKoopmanOperator_55164559950379
// MI455X (gfx1250) — compile-verified
//
#include <hip/hip_runtime.h>
#include <hip/hip_bf16.h>

// ---------------------------------------------------------------------------
// SKEL Koopman operator: out = expm(solve(N@N^T+eps, skew(R)-Q@Q^T-eps) * t) @ x
// DIM_K = 256, N_COLS = 100000, all fp32.
// Big GEMM (K @ x, 13.1 GFLOP / 205 MB) uses V_WMMA_F32_16X16X4_F32.
// ---------------------------------------------------------------------------

typedef __attribute__((ext_vector_type(2))) float v2f;
typedef __attribute__((ext_vector_type(8))) float v8f;

#define DIMK   256
#define NCOLS  100000

__device__ inline v8f wmma_f32(v2f a, v2f b, v8f c) {
  // 8 args: (neg_a, A, neg_b, B, c_mod, C, reuse_a, reuse_b)
  return __builtin_amdgcn_wmma_f32_16x16x4_f32(false, a, false, b, (short)0, c,
                                               false, false);
}

// ---------------------------------------------------------------------------
// 1) F = (R - R^T)/2 - Q@Q^T - eps*I ;  E = N@N^T + eps*I
// ---------------------------------------------------------------------------
__global__ void build_FE(const float* __restrict__ R, const float* __restrict__ Q,
                         const float* __restrict__ Nm,
                         float* __restrict__ F, float* __restrict__ E) {
  int idx = blockIdx.x * blockDim.x + threadIdx.x;   // 65536 threads
  int i = idx >> 8, j = idx & 255;
  float qq = 0.f, nn = 0.f;
  for (int k = 0; k < DIMK; ++k) {
    qq = fmaf(Q[i * DIMK + k], Q[j * DIMK + k], qq);
    nn = fmaf(Nm[i * DIMK + k], Nm[j * DIMK + k], nn);
  }
  float skew = 0.5f * (R[i * DIMK + j] - R[j * DIMK + i]);
  float eps = (i == j) ? 1e-8f : 0.f;
  F[idx] = skew - qq - eps;
  E[idx] = nn + eps;
}

// ---------------------------------------------------------------------------
// 2) Gauss-Jordan solve: X = Einp^{-1} @ Bm   (single 1024-thread workgroup)
//    W is a 256x512 global scratch ([E | B] augmented). No pivot search:
//    Einp is SPD (or a well-conditioned Pade denominator).
// ---------------------------------------------------------------------------
__global__ void gj_solve(const float* __restrict__ Einp, const float* __restrict__ Bm,
                         float* __restrict__ X, float* __restrict__ W) {
  __shared__ float fac[DIMK];
  int tid = threadIdx.x;                              // 0..1023
  for (int e = tid; e < DIMK * 512; e += 1024) {
    int i = e >> 9, j = e & 511;
    W[e] = (j < DIMK) ? Einp[i * DIMK + j] : Bm[i * DIMK + (j - DIMK)];
  }
  __threadfence_block();
  __syncthreads();
  for (int p = 0; p < DIMK; ++p) {
    float inv = 1.0f / W[p * 512 + p];                // broadcast read
    __syncthreads();                                  // read-before-write
    if (tid < 512) W[p * 512 + tid] *= inv;           // normalize pivot row
    __threadfence_block();
    __syncthreads();
    if (tid < DIMK) fac[tid] = W[tid * 512 + p];      // snapshot column p
    __syncthreads();
    for (int e = tid; e < DIMK * 512; e += 1024) {
      int i = e >> 9, j = e & 511;
      if (i != p) W[e] = fmaf(-fac[i], W[p * 512 + j], W[e]);
    }
    __threadfence_block();
    __syncthreads();
  }
  for (int e = tid; e < DIMK * DIMK; e += 1024) {
    int i = e >> 8, j = e & 255;
    X[i * DIMK + j] = W[i * 512 + DIMK + j];
  }
}

// ---------------------------------------------------------------------------
// 3) 1-norm of t*A -> squaring count s and combined scale t*2^{-s} in header
// ---------------------------------------------------------------------------
__global__ void norm_prep(const float* __restrict__ A, const float* __restrict__ t_ptr,
                          int* __restrict__ hdr) {
  __shared__ float red[DIMK];
  int j = threadIdx.x;
  float cs = 0.f;
  for (int i = 0; i < DIMK; ++i) cs += fabsf(A[i * DIMK + j]);
  red[j] = cs;
  __syncthreads();
  for (int off = 128; off > 0; off >>= 1) {
    if (j < off) red[j] = fmaxf(red[j], red[j + off]);
    __syncthreads();
  }
  if (j == 0) {
    float t = t_ptr[0];
    float nrm = red[0] * t;
    const float theta13 = 5.371920351148152f;
    int s = 0;
    if (nrm > theta13) {
      s = (int)ceilf(log2f(nrm / theta13));
      if (s < 0) s = 0;
      if (s > 32) s = 32;                              // max_squarings=32
    }
    hdr[0] = s;
    ((float*)hdr)[1] = t * exp2f(-(float)s);
  }
}

// ---------------------------------------------------------------------------
// 4) B = A * (t * 2^{-s})
// ---------------------------------------------------------------------------
__global__ void scale_mat(float* __restrict__ B, const float* __restrict__ A,
                          const int* __restrict__ hdr) {
  float sc = ((const float*)hdr)[1];
  int idx = blockIdx.x * blockDim.x + threadIdx.x;
  B[idx] = A[idx] * sc;
}

// ---------------------------------------------------------------------------
// 5) Generic 256x256 GEMM, one 16x16 tile per wave, WMMA f32 16x16x4.
// ---------------------------------------------------------------------------
__global__ void gemm256(float* __restrict__ C, const float* __restrict__ A,
                        const float* __restrict__ B) {
  int wave = threadIdx.x >> 5;
  int lane = threadIdx.x & 31;
  int tile = blockIdx.x * 8 + wave;                   // 0..255
  int tm = tile >> 4, tn = tile & 15;
  int lm = lane & 15;
  int kc = (lane >> 4) * 2;                           // 0 or 2
  v8f acc = {};
  const float* Ab = A + (tm * 16 + lm) * DIMK + kc;
  const float* Bb = B + tn * 16 + lm;
  for (int k0 = 0; k0 < DIMK; k0 += 4) {
    v2f a = *(const v2f*)(Ab + k0);
    v2f b;
    b.x = Bb[(k0 + kc) * DIMK];
    b.y = Bb[(k0 + kc + 1) * DIMK];
    acc = wmma_f32(a, b, acc);
  }
  int ro = (lane >> 4) * 8;
  for (int v = 0; v < 8; ++v)
    C[(tm * 16 + v + ro) * DIMK + tn * 16 + lm] = acc[v];
}

// ---------------------------------------------------------------------------
// 6) dst = (iter < s) ? src@src : src      (fixed 32 launches, s on device)
// ---------------------------------------------------------------------------
__global__ void square_sel(float* __restrict__ dst, const float* __restrict__ src,
                           const int* __restrict__ hdr, int iter) {
  int s = hdr[0];
  int wave = threadIdx.x >> 5;
  int lane = threadIdx.x & 31;
  int tile = blockIdx.x * 8 + wave;
  int tm = tile >> 4, tn = tile & 15;
  int lm = lane & 15;
  int kc = (lane >> 4) * 2;
  v8f acc = {};
  const float* Ab = src + (tm * 16 + lm) * DIMK + kc;
  const float* Bb = src + tn * 16 + lm;
  for (int k0 = 0; k0 < DIMK; k0 += 4) {
    v2f a = *(const v2f*)(Ab + k0);
    v2f b;
    b.x = Bb[(k0 + kc) * DIMK];
    b.y = Bb[(k0 + kc + 1) * DIMK];
    acc = wmma_f32(a, b, acc);
  }
  bool doSq = iter < s;
  int ro = (lane >> 4) * 8;
  for (int v = 0; v < 8; ++v) {
    int m = tm * 16 + v + ro, n = tn * 16 + lm;
    dst[m * DIMK + n] = doSq ? acc[v] : src[m * DIMK + n];
  }
}

// ---------------------------------------------------------------------------
// 7) Elementwise: out = cb*base + c1*X1 + c2*X2 + c3*X3 + cI*I  (null-able ptrs)
// ---------------------------------------------------------------------------
__global__ void combo(float* __restrict__ out,
                      const float* base, float cb,
                      const float* X1, float c1,
                      const float* X2, float c2,
                      const float* X3, float c3, float cI) {
  int idx = blockIdx.x * blockDim.x + threadIdx.x;
  int i = idx >> 8, j = idx & 255;
  float r = 0.f;
  if (base) r = fmaf(cb, base[idx], r);
  if (X1)   r = fmaf(c1, X1[idx], r);
  if (X2)   r = fmaf(c2, X2[idx], r);
  if (X3)   r = fmaf(c3, X3[idx], r);
  if (i == j) r += cI;
  out[idx] = r;
}

// ---------------------------------------------------------------------------
// 8) out(256 x 100000) = K(256x256) @ x(256x100000)      [the big one]
//    Block = 8 waves, owns a 32-col stripe x all 256 rows. x stripe staged
//    transposed in LDS (stride 260 => conflict-free b64 B-fragment reads).
// ---------------------------------------------------------------------------
__global__ void kx_gemm(const float* __restrict__ K, const float* __restrict__ x,
                        float* __restrict__ out) {
  __shared__ float xs[32][260];                       // xs[n][k], 33.3 KB
  int n0 = blockIdx.x * 32;
  int tid = threadIdx.x;
  for (int e = tid; e < DIMK * 32; e += 256) {        // coalesced 128B rows
    int k = e >> 5, n = e & 31;
    xs[n][k] = x[(size_t)k * NCOLS + n0 + n];
  }
  __syncthreads();

  int wave = tid >> 5;
  int lane = tid & 31;
  int m0 = wave * 32;                                 // two M-tiles per wave
  int lm = lane & 15;
  int kc = (lane >> 4) * 2;
  v8f a00 = {}, a01 = {}, a10 = {}, a11 = {};         // [mtile][ntile]
  const float* A0 = K + (m0 + lm) * DIMK + kc;
  const float* A1 = A0 + 16 * DIMK;
  for (int k0 = 0; k0 < DIMK; k0 += 4) {
    v2f fa0 = *(const v2f*)(A0 + k0);
    v2f fa1 = *(const v2f*)(A1 + k0);
    v2f fb0 = *(const v2f*)(&xs[lm][k0 + kc]);
    v2f fb1 = *(const v2f*)(&xs[lm + 16][k0 + kc]);
    a00 = wmma_f32(fa0, fb0, a00);
    a01 = wmma_f32(fa0, fb1, a01);
    a10 = wmma_f32(fa1, fb0, a10);
    a11 = wmma_f32(fa1, fb1, a11);
  }
  int ro = (lane >> 4) * 8;
  for (int v = 0; v < 8; ++v) {
    int m = m0 + v + ro;
    out[(size_t)m * NCOLS + n0 + lm]        = a00[v];
    out[(size_t)m * NCOLS + n0 + 16 + lm]   = a01[v];
    out[(size_t)(m + 16) * NCOLS + n0 + lm]      = a10[v];
    out[(size_t)(m + 16) * NCOLS + n0 + 16 + lm] = a11[v];
  }
}

// ---------------------------------------------------------------------------
extern "C" void kernel_launch(void* const* d_in, const int* in_sizes, int n_in,
                              void* d_out, int out_size, void* d_ws, size_t ws_size,
                              hipStream_t stream) {
  (void)in_sizes; (void)n_in; (void)out_size; (void)ws_size;
  const float* t  = (const float*)d_in[0];
  const float* x  = (const float*)d_in[1];
  const float* R  = (const float*)d_in[2];
  const float* Q  = (const float*)d_in[3];
  const float* Nm = (const float*)d_in[4];
  float* out = (float*)d_out;

  float* ws  = (float*)d_ws;
  int*   hdr = (int*)d_ws;
  float* base = ws + 64;
  const size_t MS = (size_t)DIMK * DIMK;              // 65536 floats per slot
  auto buf = [&](int i) { return base + (size_t)i * MS; };
  // 0:F 1:E 2:A 3:B 4:A2 5:A4 6:A6 7:T1 8:T2 9:U 10:V 11:S1 12:S2 13:K0 14:K1
  // 15..16: GJ augmented workspace (256x512).  Total ~4.5 MB of ws.

  // Pade-13 coefficients (scipy/jax expm)
  const float b0 = 6.476475253248e16f, b1 = 3.238237626624e16f,
              b2 = 7.7717703038976e15f, b3 = 1.1873537964288e15f,
              b4 = 1.29060195264e14f,  b5 = 1.05594705216e13f,
              b6 = 6.704425728e11f,    b7 = 3.352212864e10f,
              b8 = 1.32324192e9f,      b9 = 4.08408e7f,
              b10 = 960960.f, b11 = 16380.f, b12 = 182.f, b13 = 1.f;

  build_FE<<<256, 256, 0, stream>>>(R, Q, Nm, buf(0), buf(1));
  gj_solve<<<1, 1024, 0, stream>>>(buf(1), buf(0), buf(2), buf(15));   // A = E\F
  norm_prep<<<1, 256, 0, stream>>>(buf(2), t, hdr);
  scale_mat<<<256, 256, 0, stream>>>(buf(3), buf(2), hdr);             // B = A*t/2^s

  gemm256<<<32, 256, 0, stream>>>(buf(4), buf(3), buf(3));             // A2
  gemm256<<<32, 256, 0, stream>>>(buf(5), buf(4), buf(4));             // A4
  gemm256<<<32, 256, 0, stream>>>(buf(6), buf(4), buf(5));             // A6

  // U = B @ (A6@(b13*A6+b11*A4+b9*A2) + b7*A6+b5*A4+b3*A2+b1*I)
  combo<<<256, 256, 0, stream>>>(buf(7), nullptr, 0.f, buf(6), b13, buf(5), b11, buf(4), b9, 0.f);
  gemm256<<<32, 256, 0, stream>>>(buf(8), buf(6), buf(7));
  combo<<<256, 256, 0, stream>>>(buf(8), buf(8), 1.f, buf(6), b7, buf(5), b5, buf(4), b3, b1);
  gemm256<<<32, 256, 0, stream>>>(buf(9), buf(3), buf(8));
  // V = A6@(b12*A6+b10*A4+b8*A2) + b6*A6+b4*A4+b2*A2+b0*I
  combo<<<256, 256, 0, stream>>>(buf(7), nullptr, 0.f, buf(6), b12, buf(5), b10, buf(4), b8, 0.f);
  gemm256<<<32, 256, 0, stream>>>(buf(8), buf(6), buf(7));
  combo<<<256, 256, 0, stream>>>(buf(10), buf(8), 1.f, buf(6), b6, buf(5), b4, buf(4), b2, b0);

  combo<<<256, 256, 0, stream>>>(buf(11), buf(10), 1.f, buf(9), -1.f, nullptr, 0.f, nullptr, 0.f, 0.f); // V-U
  combo<<<256, 256, 0, stream>>>(buf(12), buf(10), 1.f, buf(9),  1.f, nullptr, 0.f, nullptr, 0.f, 0.f); // V+U
  gj_solve<<<1, 1024, 0, stream>>>(buf(11), buf(12), buf(13), buf(15));  // K0 = (V-U)\(V+U)

  float* cur = buf(13);
  float* nxt = buf(14);
  for (int i = 0; i < 32; ++i) {                       // conditional squarings
    square_sel<<<32, 256, 0, stream>>>(nxt, cur, hdr, i);
    float* tmp = cur; cur = nxt; nxt = tmp;
  }

  kx_gemm<<<NCOLS / 32, 256, 0, stream>>>(cur, x, out);  // 3125 blocks
}